// Coords2RMSD_28570122453898
// MI455X (gfx1250) — compile-verified
//
#include <hip/hip_runtime.h>
#include <hip/hip_bf16.h>

// Kabsch RMSD, single pass over HBM, WMMA f32 rank-4 symmetric Gram update.
// G = M M^T with M (7 x A): rows 0-2 masked src, 3-5 masked dst, 6 mask.
// rmsd = sqrt(max(0, E0 - 2*(s1+s2+sign(det H)*s3)) / n)

typedef __attribute__((ext_vector_type(2))) float v2f;
typedef __attribute__((ext_vector_type(8))) float v8f;

#define ATOMS    2048
#define ROW3     (3 * ATOMS)
#define NTHREADS 256                                     // 8 wave32
#define NWAVES   (NTHREADS / 32)
#define TPW      ((ATOMS / 4) / NWAVES)                  // 64 tiles per wave
#define LSTRIDE  2056                                    // 2048 + 8 pad: conflict-free banks

__global__ __launch_bounds__(NTHREADS) void kabsch_rmsd_kernel(
    const float* __restrict__ inp,
    const float* __restrict__ tgt,
    const long long* __restrict__ num_atoms,
    float* __restrict__ out)
{
    __shared__ __align__(16) float s_M[7 * LSTRIDE];     // ~56.2 KB

    const int b    = blockIdx.x;
    const int tid  = threadIdx.x;
    const int wave = tid >> 5;
    const int lane = tid & 31;

    const int nv = (int)num_atoms[b];

    // ---- stage masked coords into planar [row][atom] LDS layout ----
    {
        const float* gs = inp + (size_t)b * ROW3;
        const float* gd = tgt + (size_t)b * ROW3;
        #pragma unroll
        for (int a = tid; a < ATOMS; a += NTHREADS) {
            const float m  = (a < nv) ? 1.0f : 0.0f;
            const float sx = gs[3 * a + 0], sy = gs[3 * a + 1], sz = gs[3 * a + 2];
            const float dx = gd[3 * a + 0], dy = gd[3 * a + 1], dz = gd[3 * a + 2];
            s_M[0 * LSTRIDE + a] = sx * m;
            s_M[1 * LSTRIDE + a] = sy * m;
            s_M[2 * LSTRIDE + a] = sz * m;
            s_M[3 * LSTRIDE + a] = dx * m;
            s_M[4 * LSTRIDE + a] = dy * m;
            s_M[5 * LSTRIDE + a] = dz * m;
            s_M[6 * LSTRIDE + a] = m;
        }
    }
    __syncthreads();

    // ---- WMMA rank-4 Gram accumulation ----
    // Operand element (lane, vgpr v) = M[lane&15][4*t + 2*(lane>>4) + v].
    // Lanes with row >= 7 alias row 6 (broadcast read); they only pollute
    // G rows/cols >= 7, which are never read back.
    // Uniform constant trip count so the backedge stays scalar and unrolling
    // batches the ds_load_b64s ahead of the WMMA dependency chain.
    const int r  = lane & 15;
    const int kh = lane >> 4;
    const int rr = (r < 7) ? r : 6;
    const float2* p =
        (const float2*)(s_M + rr * LSTRIDE) + wave * (TPW * 2) + kh;

    v8f c = {0.f, 0.f, 0.f, 0.f, 0.f, 0.f, 0.f, 0.f};

    #pragma unroll 16
    for (int i = 0; i < TPW; ++i) {
        const float2 mv = p[i * 2];                      // ds_load_b64, +16B steps
        v2f av;
        av.x = mv.x;
        av.y = mv.y;
        c = __builtin_amdgcn_wmma_f32_16x16x4_f32(
            false, av, false, av, (short)0, c, false, false);
    }

    // ---- all LDS reads done; reuse s_M[0..48] as reduction scratch ----
    __syncthreads();
    if (tid < 49) s_M[tid] = 0.0f;
    __syncthreads();

    // C vgpr m, lanes 0..15 hold G[m][lane]; we need the 7x7 corner.
    union { v8f v; float f[8]; } cu;
    cu.v = c;
    if (lane < 7) {
        #pragma unroll
        for (int m = 0; m < 7; ++m)
            atomicAdd(&s_M[m * 7 + lane], cu.f[m]);      // ds_add_f32
    }
    __syncthreads();

    // ---- scalar f64 epilogue: closed-form Kabsch residual ----
    if (tid == 0) {
        double G[7][7];
        #pragma unroll
        for (int i = 0; i < 7; ++i)
            #pragma unroll
            for (int j = 0; j < 7; ++j)
                G[i][j] = (double)s_M[i * 7 + j];

        const double n = (double)nv;
        double S[3] = {G[6][0], G[6][1], G[6][2]};
        double D[3] = {G[6][3], G[6][4], G[6][5]};
        double ss = G[0][0] + G[1][1] + G[2][2];
        double dd = G[3][3] + G[4][4] + G[5][5];
        double E0 = (ss - (S[0]*S[0] + S[1]*S[1] + S[2]*S[2]) / n)
                  + (dd - (D[0]*D[0] + D[1]*D[1] + D[2]*D[2]) / n);

        double H[3][3];
        #pragma unroll
        for (int i = 0; i < 3; ++i)
            #pragma unroll
            for (int j = 0; j < 3; ++j)
                H[i][j] = G[i][3 + j] - S[i] * D[j] / n;

        double det =
            H[0][0] * (H[1][1] * H[2][2] - H[1][2] * H[2][1])
          - H[0][1] * (H[1][0] * H[2][2] - H[1][2] * H[2][0])
          + H[0][2] * (H[1][0] * H[2][1] - H[1][1] * H[2][0]);

        double K00 = H[0][0]*H[0][0] + H[1][0]*H[1][0] + H[2][0]*H[2][0];
        double K11 = H[0][1]*H[0][1] + H[1][1]*H[1][1] + H[2][1]*H[2][1];
        double K22 = H[0][2]*H[0][2] + H[1][2]*H[1][2] + H[2][2]*H[2][2];
        double K01 = H[0][0]*H[0][1] + H[1][0]*H[1][1] + H[2][0]*H[2][1];
        double K02 = H[0][0]*H[0][2] + H[1][0]*H[1][2] + H[2][0]*H[2][2];
        double K12 = H[0][1]*H[0][2] + H[1][1]*H[1][2] + H[2][1]*H[2][2];

        double q  = (K00 + K11 + K22) * (1.0 / 3.0);
        double p1 = K01*K01 + K02*K02 + K12*K12;
        double p2 = (K00-q)*(K00-q) + (K11-q)*(K11-q) + (K22-q)*(K22-q) + 2.0*p1;
        double l1, l2, l3;
        if (p2 < 1e-300) {
            l1 = l2 = l3 = q;
        } else {
            double p_  = sqrt(p2 / 6.0);
            double ip  = 1.0 / p_;
            double B00 = (K00 - q) * ip, B11 = (K11 - q) * ip, B22 = (K22 - q) * ip;
            double B01 = K01 * ip, B02 = K02 * ip, B12 = K12 * ip;
            double detB =
                B00 * (B11 * B22 - B12 * B12)
              - B01 * (B01 * B22 - B12 * B02)
              + B02 * (B01 * B12 - B11 * B02);
            double rr2 = fmin(1.0, fmax(-1.0, 0.5 * detB));
            double phi = acos(rr2) * (1.0 / 3.0);
            l1 = q + 2.0 * p_ * cos(phi);
            l3 = q + 2.0 * p_ * cos(phi + 2.0943951023931953);
            l2 = 3.0 * q - l1 - l3;
        }
        double s1 = sqrt(fmax(l1, 0.0));
        double s2 = sqrt(fmax(l2, 0.0));
        double s3 = sqrt(fmax(l3, 0.0));
        double sgn = (det > 0.0) ? 1.0 : ((det < 0.0) ? -1.0 : 0.0);
        double E = s1 + s2 + sgn * s3;

        double msd = fmax(0.0, E0 - 2.0 * E) / n;
        out[b] = (float)sqrt(msd);
    }
}

extern "C" void kernel_launch(void* const* d_in, const int* in_sizes, int n_in,
                              void* d_out, int out_size, void* d_ws, size_t ws_size,
                              hipStream_t stream) {
    const float*     inp = (const float*)d_in[0];
    const float*     tgt = (const float*)d_in[1];
    const long long* na  = (const long long*)d_in[2];
    float*           out = (float*)d_out;

    const int B = in_sizes[2];  // 4096

    kabsch_rmsd_kernel<<<dim3(B), dim3(NTHREADS), 0, stream>>>(inp, tgt, na, out);
}